// ConditionalAdjunctionModelV3_78142634983724
// MI455X (gfx1250) — compile-verified
//
#include <hip/hip_runtime.h>
#include <math.h>

typedef float v2f __attribute__((ext_vector_type(2)));
typedef float v8f __attribute__((ext_vector_type(8)));

#define NB   8
#define NPTS 16384
#define NM   512
#define NA   5
#define EPSD 1e-12f
#define FLTMAX 3.402823466e+38f

// Workspace layout (floats):
//   [0, NB*NM)               backward per-(b,j) running SQUARED mins
//   [NB*NM, NB*NM+NB)        forward per-batch sums
//   [NB*NM+NB, +NB*NA)       affordance per-batch sums
#define WS_BACK 0
#define WS_FWD  (NB*NM)
#define WS_AFF  (NB*NM + NB)
#define WS_TOT  (NB*NM + NB + NB*NA)

__global__ __launch_bounds__(256) void init_ws_kernel(float* __restrict__ ws) {
    int i = blockIdx.x * 256 + threadIdx.x;
    if (i < NB*NM)       ws[i] = FLTMAX;
    else if (i < WS_TOT) ws[i] = 0.0f;
}

__global__ __launch_bounds__(256) void chamfer_wmma_kernel(
        const float* __restrict__ pos,     // (NB*NPTS, 3)
        const float* __restrict__ recon,   // (NB, NM, 3)
        float* __restrict__ out,           // full output buffer
        float* __restrict__ ws) {
    __shared__ float s_recon[NM * 3];
    __shared__ float s_yn2[NM];

    const int tid  = threadIdx.x;
    const int wave = tid >> 5;
    const int lane = tid & 31;
    const int half = lane >> 4;   // 0 or 1
    const int li   = lane & 15;

    const int chunk  = blockIdx.x * 8 + wave;  // 16-point chunk of originals
    const int b      = chunk >> 10;            // 1024 chunks per batch
    const int c_in_b = chunk & 1023;

    // Stage this batch's reconstructed points (+ squared norms) in LDS.
    // All 8 waves of a block share the same batch (1024 % 8 == 0).
    for (int i = tid; i < NM * 3; i += 256)
        s_recon[i] = recon[(size_t)b * NM * 3 + i];
    __syncthreads();
    for (int j = tid; j < NM; j += 256) {
        float qx = s_recon[j*3+0], qy = s_recon[j*3+1], qz = s_recon[j*3+2];
        s_yn2[j] = qx*qx + qy*qy + qz*qz;
    }
    __syncthreads();

    // ---- A matrix: 16 original points, 16x4 f32 (K padded 3 -> 4) ----
    // Lane l<16 : v0 = A[M=l][K=0]=x , v1 = A[M=l][K=1]=y
    // Lane l>=16: v0 = A[M=l-16][K=2]=z, v1 = A[M=l-16][K=3]=0
    const int    pbase = b * NPTS + c_in_b * 16;
    const size_t poff  = (size_t)(pbase + li) * 3;
    float px = pos[poff + 0], py = pos[poff + 1], pz = pos[poff + 2];
    float xn2 = px*px + py*py + pz*pz;

    v2f A;
    A.x = half ? pz : px;
    A.y = half ? 0.0f : py;

    // Broadcast |x|^2 for the 8 rows this half owns (m = v + 8*half).
    float xn2m[8];
    #pragma unroll
    for (int v = 0; v < 8; ++v)
        xn2m[v] = __shfl(xn2, v + 8 * half, 32);

    // Track SQUARED distance mins in the loop; sqrt is monotone so
    // min(sqrt(max(d2,eps))) == sqrt(max(min(d2),eps)) — one sqrt at the end.
    float rowmin2[8];
    #pragma unroll
    for (int v = 0; v < 8; ++v) rowmin2[v] = FLTMAX;

    for (int t = 0; t < NM / 16; ++t) {
        const int j = t * 16 + li;
        float qx = s_recon[j*3+0], qy = s_recon[j*3+1], qz = s_recon[j*3+2];
        float yn2 = s_yn2[j];

        // B matrix 4x16 f32: lane l<16: v0=B[0][l]=x, v1=B[1][l]=y
        //                    lane l>=16: v0=B[2][l-16]=z, v1=0
        v2f Bm;
        Bm.x = half ? qz : qx;
        Bm.y = half ? 0.0f : qy;

        v8f C = {};
        C = __builtin_amdgcn_wmma_f32_16x16x4_f32(
                /*neg_a=*/false, A, /*neg_b=*/false, Bm,
                /*c_mod=*/(short)0, C, /*reuse_a=*/false, /*reuse_b=*/false);

        // D[v] @ lane = dot(orig m = v+8*half, recon n = t*16+li)
        float colmin2 = FLTMAX;
        #pragma unroll
        for (int v = 0; v < 8; ++v) {
            float d2 = (xn2m[v] + yn2) - 2.0f * C[v];
            rowmin2[v] = fminf(rowmin2[v], d2);
            colmin2    = fminf(colmin2, d2);
        }
        // combine the two halves (rows 0..7 with rows 8..15) for column j
        colmin2 = fminf(colmin2, __shfl_xor(colmin2, 16, 32));
        if (half == 0) {
            // non-negative floats: int bit-pattern compare preserves order
            atomicMin((int*)&ws[WS_BACK + b * NM + j], __float_as_int(colmin2));
        }
    }

    // Reduce row mins across the 16 lanes of each half (covers all 512 cols).
    #pragma unroll
    for (int v = 0; v < 8; ++v) {
        rowmin2[v] = fminf(rowmin2[v], __shfl_xor(rowmin2[v], 1, 32));
        rowmin2[v] = fminf(rowmin2[v], __shfl_xor(rowmin2[v], 2, 32));
        rowmin2[v] = fminf(rowmin2[v], __shfl_xor(rowmin2[v], 4, 32));
        rowmin2[v] = fminf(rowmin2[v], __shfl_xor(rowmin2[v], 8, 32));
    }

    // One sqrt per original point (v_sqrt_f32, no software fixup ladder).
    float pp[8];
    #pragma unroll
    for (int v = 0; v < 8; ++v)
        pp[v] = __builtin_amdgcn_sqrtf(fmaxf(rowmin2[v], EPSD));

    // coherence_spatial lives at out + NB; per_point index = b*NPTS + c_in_b*16 + m
    float* spatial = out + NB;
    #pragma unroll
    for (int v = 0; v < 8; ++v) {
        if (li == v) {
            int m = v + 8 * half;
            spatial[(size_t)b * NPTS + c_in_b * 16 + m] = pp[v];
        }
    }

    // forward sum: lane 0 covers rows 0..7, lane 16 covers rows 8..15
    if (li == 0) {
        float s = 0.0f;
        #pragma unroll
        for (int v = 0; v < 8; ++v) s += pp[v];
        atomicAdd(&ws[WS_FWD + b], s);
    }
}

__global__ __launch_bounds__(256) void aff_reduce_kernel(
        const float* __restrict__ aff, float* __restrict__ ws) {
    const int b = blockIdx.x;
    const int tid = threadIdx.x;
    float acc[NA] = {0.f, 0.f, 0.f, 0.f, 0.f};
    for (int i = tid; i < NPTS; i += 256) {
        size_t row = ((size_t)b * NPTS + i) * NA;
        #pragma unroll
        for (int a = 0; a < NA; ++a) acc[a] += aff[row + a];
    }
    #pragma unroll
    for (int a = 0; a < NA; ++a)
        atomicAdd(&ws[WS_AFF + b * NA + a], acc[a]);
}

__global__ __launch_bounds__(256) void finalize_kernel(
        const float* __restrict__ ws, float* __restrict__ out) {
    __shared__ float sdata[256];
    const int b = blockIdx.x;
    const int tid = threadIdx.x;

    // backward mins are stored squared: sqrt(max(d2, eps)) here
    float s0 = __builtin_amdgcn_sqrtf(fmaxf(ws[WS_BACK + b * NM + tid], EPSD));
    float s1 = __builtin_amdgcn_sqrtf(fmaxf(ws[WS_BACK + b * NM + 256 + tid], EPSD));
    sdata[tid] = s0 + s1;
    __syncthreads();
    for (int off = 128; off > 0; off >>= 1) {
        if (tid < off) sdata[tid] += sdata[tid + off];
        __syncthreads();
    }
    if (tid == 0) {
        float bwd = sdata[0] / (float)NM;
        float fwd = ws[WS_FWD + b] / (float)NPTS;
        out[b] = 0.5f * (fwd + bwd);                  // coherence_scalar
    }
    if (tid < NA) {
        out[NB + (size_t)NB * NPTS + b * NA + tid] =
            ws[WS_AFF + b * NA + tid] / (float)NPTS;  // affordances_batched
    }
}

extern "C" void kernel_launch(void* const* d_in, const int* in_sizes, int n_in,
                              void* d_out, int out_size, void* d_ws, size_t ws_size,
                              hipStream_t stream) {
    const float* pos   = (const float*)d_in[0];  // (NB*NPTS, 3) f32
    const float* recon = (const float*)d_in[1];  // (NB, NM, 3)  f32
    const float* aff   = (const float*)d_in[2];  // (NB*NPTS, NA) f32
    // d_in[3] = batch (int64) is implicit: batch[i] = i / NPTS
    float* out = (float*)d_out;
    float* ws  = (float*)d_ws;

    init_ws_kernel<<<(WS_TOT + 255) / 256, 256, 0, stream>>>(ws);
    // 8192 chunks of 16 originals, 8 waves (256 threads) per block
    chamfer_wmma_kernel<<<(NB * NPTS / 16) / 8, 256, 0, stream>>>(pos, recon, out, ws);
    aff_reduce_kernel<<<NB, 256, 0, stream>>>(aff, ws);
    finalize_kernel<<<NB, 256, 0, stream>>>(ws, out);
}